// SelfNonParametric_Mod1_70531952935516
// MI455X (gfx1250) — compile-verified
//
#include <hip/hip_runtime.h>
#include <hip/hip_bf16.h>

#define N_LB   1024
#define N_ULB  7168
#define NTOT   8192
#define DFEAT  128
#define NC     10
#define P_CUT  0.95f
#define ATT_CUT 0.6f

typedef float v2f __attribute__((ext_vector_type(2)));
typedef float v8f __attribute__((ext_vector_type(8)));

#define NEG_BIG (-1.0e30f)

__device__ __forceinline__ float fexp2(float x) { return __builtin_amdgcn_exp2f(x); }

// ---------------------------------------------------------------------------
// class_num[c] = sum_i lb_one_hot[i][c]   (base before histogram atomics)
// ---------------------------------------------------------------------------
__global__ void k_init_classnum(const float* __restrict__ lb_one_hot,
                                float* __restrict__ class_num) {
    int c = threadIdx.x;
    if (c < NC) {
        float s = 0.f;
        for (int i = 0; i < N_LB; ++i) s += lb_one_hot[i * NC + c];
        class_num[c] = s;
    }
}

// ---------------------------------------------------------------------------
// Per ulb row: softmax(logits1) -> class_val[1024+i]; pseudo-label histogram
// ---------------------------------------------------------------------------
__global__ void k_ulb(const float* __restrict__ l1g,
                      const float* __restrict__ l2g,
                      float* __restrict__ class_val,
                      float* __restrict__ class_num) {
    int i = blockIdx.x * blockDim.x + threadIdx.x;
    if (i >= N_ULB) return;

    float l1[NC], l2[NC];
#pragma unroll
    for (int c = 0; c < NC; ++c) { l1[c] = l1g[i*NC+c]; l2[c] = l2g[i*NC+c]; }

    float m1 = l1[0], m2 = l2[0];
#pragma unroll
    for (int c = 1; c < NC; ++c) { m1 = fmaxf(m1, l1[c]); m2 = fmaxf(m2, l2[c]); }

    float z1 = 0.f, zt1 = 0.f, zt2 = 0.f;
#pragma unroll
    for (int c = 0; c < NC; ++c) {
        z1  += __expf(l1[c] - m1);
        zt1 += __expf(2.f * (l1[c] - m1));   // temp 0.5 softmax partition
        zt2 += __expf(2.f * (l2[c] - m2));
    }
#pragma unroll
    for (int c = 0; c < NC; ++c)
        class_val[(N_LB + i) * NC + c] = __expf(l1[c] - m1) / z1;

    // mp1 >= mp2  <=>  1/zt1 >= 1/zt2  <=>  zt1 <= zt2
    bool gt = (zt1 <= zt2);
    float lg[NC]; float mg = NEG_BIG;
#pragma unroll
    for (int c = 0; c < NC; ++c) { lg[c] = gt ? l1[c] : l2[c]; mg = fmaxf(mg, lg[c]); }
    float zg = 0.f;
#pragma unroll
    for (int c = 0; c < NC; ++c) zg += __expf(lg[c] - mg);

    float maxv = NEG_BIG; int idx = 0;
#pragma unroll
    for (int c = 0; c < NC; ++c) {
        float p = __expf(lg[c] - mg) / zg;
        float v = (p >= P_CUT) ? lg[c] : 0.f;
        if (v > maxv) { maxv = v; idx = c; }
    }
    if (maxv != 0.f) atomicAdd(&class_num[idx], 1.0f);
}

// ---------------------------------------------------------------------------
// Fused attention with async-to-LDS double buffering.
// Block = 1 row tile (16 rows) x 4 waves; wave w owns j-tiles [w*128, w*128+128).
// B tile (16 rows x 512B, contiguous) staged via GLOBAL_LOAD_ASYNC_TO_LDS_B128
// into a wave-private double buffer; XOR swizzle (off ^ (row<<4)) makes the
// row-strided ds_load_b64 reads bank-conflict free. B fragments are register
// double-buffered in groups of 4 k-chunks so DS waits don't drain (s_wait_dscnt
// can stay at 8 outstanding while 8 WMMAs execute).
// ---------------------------------------------------------------------------
#define WAVES_PB 4
#define JT_PER_WAVE (NTOT / 16 / WAVES_PB)   // 128
#define TILE_BYTES 8192                      // 16 rows * 128 floats * 4B

__device__ __forceinline__ void async_tile(const char* gbase, unsigned ldsbuf, int lane) {
#pragma unroll
    for (int r = 0; r < 16; ++r) {
        unsigned long long ga = (unsigned long long)(size_t)gbase
                                + (unsigned)(r * 512 + lane * 16);
        unsigned la = ldsbuf + (unsigned)(r * 512)
                      + (((unsigned)lane * 16u) ^ (unsigned)(r << 4));
        asm volatile("global_load_async_to_lds_b128 %0, %1, off"
                     :: "v"(la), "v"(ga) : "memory");
    }
}

__global__ __launch_bounds__(32 * WAVES_PB)
void k_attn(const float* __restrict__ lb_feat,
            const float* __restrict__ anchor,
            const float* __restrict__ class_val,
            float* __restrict__ out_lb,
            float* __restrict__ out_ulb) {
    extern __shared__ char smem[];                  // WAVES_PB * 2 * 8KB
    __shared__ float sM[WAVES_PB * 16];
    __shared__ float sZ[WAVES_PB * 16];
    __shared__ int   sJ[WAVES_PB * 16];

    const int lane  = threadIdx.x & 31;
    const int wave  = threadIdx.x >> 5;
    const int rtile = blockIdx.x;                   // 0..511
    const int m     = lane & 15;
    const int kh    = lane >> 4;
    const int row0  = rtile * 16;

    const float SCALE = 14.426950408889634f;        // log2(e) / 0.1

    // ---- preload & prescale A fragments: 16x128 -> 32 chunks of 16x4 ----
    const int arow = row0 + m;
    const float* aptr = (arow < N_LB) ? (lb_feat + (size_t)arow * DFEAT)
                                      : (anchor  + (size_t)(arow - N_LB) * DFEAT);
    v2f a[32];
#pragma unroll
    for (int kk = 0; kk < 32; ++kk) {
        a[kk].x = aptr[kk * 4 + 2 * kh]     * SCALE;
        a[kk].y = aptr[kk * 4 + 2 * kh + 1] * SCALE;
    }

    // LDS base (32-bit LDS offset) for this wave's double buffer
    unsigned lds_base = (unsigned)(size_t)(__attribute__((address_space(3))) char*)smem
                        + (unsigned)(wave * 2 * TILE_BYTES);
    const char* bufgen = smem + wave * 2 * TILE_BYTES;   // generic view for reads

    // per-lane online softmax state (slot p -> row p + 8*kh)
    float M[8], Z[8]; int bj[8];
#pragma unroll
    for (int p = 0; p < 8; ++p) { M[p] = NEG_BIG; Z[p] = 0.f; bj[p] = 0; }

    const int jt0 = wave * JT_PER_WAVE;
    // prefetch first tile
    {
        int c0 = jt0 * 16;
        const char* gb = (const char*)((c0 < N_LB) ? (lb_feat + (size_t)c0 * DFEAT)
                                                   : (anchor  + (size_t)(c0 - N_LB) * DFEAT));
        async_tile(gb, lds_base, lane);
    }

    const unsigned mx   = (unsigned)(m << 4);
    const unsigned rowb = (unsigned)(m * 512 + kh * 8);

    for (int t = 0; t < JT_PER_WAVE; ++t) {
        if (t + 1 < JT_PER_WAVE) {
            int c0 = (jt0 + t + 1) * 16;
            const char* gb = (const char*)((c0 < N_LB) ? (lb_feat + (size_t)c0 * DFEAT)
                                                       : (anchor  + (size_t)(c0 - N_LB) * DFEAT));
            async_tile(gb, lds_base + (unsigned)(((t + 1) & 1) * TILE_BYTES), lane);
            asm volatile("s_wait_asynccnt 0x10" ::: "memory");   // current tile landed
        } else {
            asm volatile("s_wait_asynccnt 0x0" ::: "memory");
        }

        const char* bufp = bufgen + (t & 1) * TILE_BYTES;
        v8f c0v = {0.f,0.f,0.f,0.f,0.f,0.f,0.f,0.f};
        v8f c1v = {0.f,0.f,0.f,0.f,0.f,0.f,0.f,0.f};

        // register double-buffered B fragments: groups of 4 k-chunks
        v2f b0[2][4], b1[2][4];
#pragma unroll
        for (int q = 0; q < 4; ++q) {
            b0[0][q] = *(const v2f*)(bufp + rowb + (((unsigned)(q << 4)) ^ mx));
            b1[0][q] = *(const v2f*)(bufp + rowb + (((unsigned)((q + 16) << 4)) ^ mx));
        }
#pragma unroll
        for (int g = 0; g < 4; ++g) {
            if (g < 3) {
#pragma unroll
                for (int q = 0; q < 4; ++q) {
                    int kk = (g + 1) * 4 + q;
                    b0[(g + 1) & 1][q] = *(const v2f*)(bufp + rowb + (((unsigned)(kk << 4)) ^ mx));
                    b1[(g + 1) & 1][q] = *(const v2f*)(bufp + rowb + (((unsigned)((kk + 16) << 4)) ^ mx));
                }
            }
#pragma unroll
            for (int q = 0; q < 4; ++q) {
                int kk = g * 4 + q;
                c0v = __builtin_amdgcn_wmma_f32_16x16x4_f32(false, a[kk],      false, b0[g & 1][q],
                                                            (short)0, c0v, false, false);
                c1v = __builtin_amdgcn_wmma_f32_16x16x4_f32(false, a[kk + 16], false, b1[g & 1][q],
                                                            (short)0, c1v, false, false);
            }
        }
        v8f cs = c0v + c1v;

        const int j = (jt0 + t) * 16 + m;
#pragma unroll
        for (int p = 0; p < 8; ++p) {
            float v  = cs[p];
            float mo = M[p];
            float mn = fmaxf(mo, v);
            Z[p]  = Z[p] * fexp2(mo - mn) + fexp2(v - mn);
            bj[p] = (v > mo) ? j : bj[p];
            M[p]  = mn;
        }
    }

    // ---- butterfly allreduce over the 16 lanes of each half ----
#pragma unroll
    for (int off = 8; off >= 1; off >>= 1) {
#pragma unroll
        for (int p = 0; p < 8; ++p) {
            float Mo = __shfl_xor(M[p], off);
            float Zo = __shfl_xor(Z[p], off);
            int   jo = __shfl_xor(bj[p], off);
            float mn = fmaxf(M[p], Mo);
            Z[p]  = Z[p] * fexp2(M[p] - mn) + Zo * fexp2(Mo - mn);
            bj[p] = (Mo > M[p]) ? jo : bj[p];
            M[p]  = mn;
        }
    }

    // ---- publish per-wave partials, combine across the 4 j-chunks ----
    if (m == 0) {
#pragma unroll
        for (int p = 0; p < 8; ++p) {
            int row = p + 8 * kh;
            sM[wave * 16 + row] = M[p];
            sZ[wave * 16 + row] = Z[p];
            sJ[wave * 16 + row] = bj[p];
        }
    }
    __syncthreads();

    if (threadIdx.x < 16) {
        int row = threadIdx.x;
        float Mv = sM[row]; float Zv = sZ[row]; int jv = sJ[row];
#pragma unroll
        for (int w = 1; w < WAVES_PB; ++w) {
            float Mo = sM[w * 16 + row];
            float Zo = sZ[w * 16 + row];
            int   jo = sJ[w * 16 + row];
            float mn = fmaxf(Mv, Mo);
            Zv = Zv * fexp2(Mv - mn) + Zo * fexp2(Mo - mn);
            jv = (Mo > Mv) ? jo : jv;
            Mv = mn;
        }
        float att = 1.0f / Zv;                      // softmax value at argmax
        int r = row0 + row;
        float* outp = (r < N_LB) ? (out_lb + (size_t)r * NC)
                                 : (out_ulb + (size_t)(r - N_LB) * NC);
        if (att >= ATT_CUT) {
            const float* cv = class_val + (size_t)jv * NC;
#pragma unroll
            for (int c = 0; c < NC; ++c) outp[c] = att * cv[c];
        } else {
#pragma unroll
            for (int c = 0; c < NC; ++c) outp[c] = 0.f;
        }
    }
}

// ---------------------------------------------------------------------------
extern "C" void kernel_launch(void* const* d_in, const int* in_sizes, int n_in,
                              void* d_out, int out_size, void* d_ws, size_t ws_size,
                              hipStream_t stream) {
    const float* anchor   = (const float*)d_in[0];  // 7168*128
    const float* positive = (const float*)d_in[1];  // 7168*128
    const float* lb_feat  = (const float*)d_in[2];  // 1024*128
    const float* lb_1hot  = (const float*)d_in[3];  // 1024*10
    const float* logits1  = (const float*)d_in[5];  // 7168*10
    const float* logits2  = (const float*)d_in[6];  // 7168*10

    float* out = (float*)d_out;
    const size_t o_anchor = 0;
    const size_t o_pos    = o_anchor + (size_t)N_ULB * DFEAT;   //  917504
    const size_t o_lbf    = o_pos    + (size_t)N_ULB * DFEAT;   // 1835008
    const size_t o_1hot   = o_lbf    + (size_t)N_LB  * DFEAT;   // 1966080
    const size_t o_lb_new = o_1hot   + (size_t)N_LB  * NC;      // 1976320
    const size_t o_u1_new = o_lb_new + (size_t)N_LB  * NC;      // 1986560
    const size_t o_u2     = o_u1_new + (size_t)N_ULB * NC;      // 2058240
    const size_t o_cnum   = o_u2     + (size_t)N_ULB * NC;      // 2129920

    float* class_val = (float*)d_ws;                 // 8192 x 10 floats

    // pass-through outputs (device-to-device, capture-safe)
    (void)hipMemcpyAsync(out + o_anchor, anchor,   (size_t)N_ULB * DFEAT * 4, hipMemcpyDeviceToDevice, stream);
    (void)hipMemcpyAsync(out + o_pos,    positive, (size_t)N_ULB * DFEAT * 4, hipMemcpyDeviceToDevice, stream);
    (void)hipMemcpyAsync(out + o_lbf,    lb_feat,  (size_t)N_LB  * DFEAT * 4, hipMemcpyDeviceToDevice, stream);
    (void)hipMemcpyAsync(out + o_1hot,   lb_1hot,  (size_t)N_LB  * NC    * 4, hipMemcpyDeviceToDevice, stream);
    (void)hipMemcpyAsync(out + o_u2,     logits2,  (size_t)N_ULB * NC    * 4, hipMemcpyDeviceToDevice, stream);
    // class_val rows 0..1023 = lb_one_hot
    (void)hipMemcpyAsync(class_val, lb_1hot, (size_t)N_LB * NC * 4, hipMemcpyDeviceToDevice, stream);

    k_init_classnum<<<1, 32, 0, stream>>>(lb_1hot, out + o_cnum);
    k_ulb<<<(N_ULB + 127) / 128, 128, 0, stream>>>(logits1, logits2, class_val, out + o_cnum);

    const size_t dyn_lds = (size_t)WAVES_PB * 2 * TILE_BYTES;   // 64 KB
    k_attn<<<NTOT / 16, 32 * WAVES_PB, dyn_lds, stream>>>(
        lb_feat, anchor, class_val, out + o_lb_new, out + o_u1_new);
}